// EnhancedGIN_37881611551313
// MI455X (gfx1250) — compile-verified
//
#include <hip/hip_runtime.h>
#include <math.h>

typedef __attribute__((ext_vector_type(2))) float v2f;
typedef __attribute__((ext_vector_type(8))) float v8f;

#define EPSN 1e-5f
#define BKC  64      // K-chunk staged in LDS per iteration
#define LDSP 288     // floats per k-PAIR row in LDS (128*2 data + 32 pad -> bank-conflict-free)

__device__ __forceinline__ float gelu_f(float x) {
    // exact erf gelu (matches torch F.gelu default / jax approximate=False)
    return 0.5f * x * (1.0f + erff(x * 0.7071067811865476f));
}

__device__ __forceinline__ float wave_sum32(float v) {
    #pragma unroll
    for (int o = 16; o > 0; o >>= 1) v += __shfl_xor(v, o, 32);
    return v;
}

// ---------------------------------------------------------------------------
// WMMA fp32 GEMM: C[n,M] = act(A[n,K] @ W[K,M] + bias)
// Block = 256 threads (8 waves), block tile 128x128.
// Each wave computes a 16x128 strip: one A fragment feeds 8 WMMAs.
// B is staged in LDS (k-pair interleaved + padded) -> ds_load_b64 fragments.
// Requires K % 64 == 0 and M % 128 == 0 (true for all GEMMs here).
// act: 0 = none, 1 = gelu, 2 = sigmoid
// ---------------------------------------------------------------------------
__global__ __launch_bounds__(256) void gemm_f32_wmma(
        const float* __restrict__ A, const float* __restrict__ W,
        const float* __restrict__ bias, float* __restrict__ C,
        int n, int K, int M, int act) {
    __shared__ float Bs[(BKC / 2) * LDSP];   // 32 * 288 * 4B = 36864 B

    const int tid  = threadIdx.x;
    const int lane = tid & 31;
    const int wid  = tid >> 5;               // 0..7 -> M subtile of the block
    const int l16  = lane & 15;
    const int hi   = lane >> 4;              // 0: lanes 0-15, 1: lanes 16-31
    const int rowbase = blockIdx.x * 128;
    const int colbase = blockIdx.y * 128;
    const int row  = rowbase + wid * 16 + l16;
    const int rowc = row < n ? row : (n - 1);            // clamp: loads always safe
    const float* __restrict__ Arow = A + (size_t)rowc * K;

    v8f acc[8];
    #pragma unroll
    for (int s = 0; s < 8; ++s) acc[s] = (v8f){0.f,0.f,0.f,0.f,0.f,0.f,0.f,0.f};

    for (int k0 = 0; k0 < K; k0 += BKC) {
        __syncthreads();
        // cooperative load: B[k0..k0+63][colbase..colbase+127] -> LDS,
        // element B[r][c] stored at (r>>1)*LDSP + 2*c + (r&1)
        #pragma unroll
        for (int j = 0; j < 16; ++j) {                   // 4096 float2 / 256 threads
            const int idx2 = j * 256 + tid;
            const int r = idx2 >> 6;                     // 64 float2 per 128-col row
            const int c = (idx2 & 63) * 2;
            const float2 v = *(const float2*)(W + (size_t)(k0 + r) * M + colbase + c);
            float* dstp = Bs + (r >> 1) * LDSP + (r & 1);
            dstp[2 * c]       = v.x;
            dstp[2 * (c + 1)] = v.y;
        }
        __syncthreads();

        #pragma unroll 4
        for (int kk = 0; kk < BKC; kk += 4) {
            const int ka = k0 + kk + 2 * hi;             // A/B k-row for this half-wave
            v2f a;
            a.x = Arow[ka];
            a.y = Arow[ka + 1];
            const float* bp = Bs + ((kk >> 1) + hi) * LDSP + 2 * l16;
            #pragma unroll
            for (int s = 0; s < 8; ++s) {
                v2f b;
                b.x = bp[s * 32];                        // B[ka  ][col0 + s*16 + l16]
                b.y = bp[s * 32 + 1];                    // B[ka+1][col0 + s*16 + l16]
                acc[s] = __builtin_amdgcn_wmma_f32_16x16x4_f32(
                    false, a, false, b, (short)0, acc[s], false, false);
            }
        }
    }

    // C/D layout: vgpr r, lanes 0-15 -> (M=r, N=lane); lanes 16-31 -> (M=r+8, N=lane-16)
    #pragma unroll
    for (int s = 0; s < 8; ++s) {
        const int col = colbase + s * 16 + l16;
        const float bv = bias ? bias[col] : 0.f;
        #pragma unroll
        for (int r = 0; r < 8; ++r) {
            const int rr = rowbase + wid * 16 + r + 8 * hi;
            if (rr < n) {
                float v = acc[s][r] + bv;
                if (act == 1)      v = gelu_f(v);
                else if (act == 2) v = 1.f / (1.f + expf(-v));
                C[(size_t)rr * M + col] = v;
            }
        }
    }
}

// ---------------------------------------------------------------------------
// small utility kernels
// ---------------------------------------------------------------------------
__global__ void zero_f(float* __restrict__ p, size_t n) {
    for (size_t i = (size_t)blockIdx.x * blockDim.x + threadIdx.x; i < n;
         i += (size_t)gridDim.x * blockDim.x) p[i] = 0.f;
}

// column (per-feature) sum and sumsq over rows, via per-block partials + atomics
__global__ void col_stats(const float* __restrict__ X, int n, int H,
                          float* __restrict__ sum, float* __restrict__ sumsq) {
    const int t = threadIdx.x;   // blockDim.x == H
    float s = 0.f, ss = 0.f;
    for (int r = blockIdx.x; r < n; r += gridDim.x) {
        float v = X[(size_t)r * H + t];
        s += v; ss += v * v;
    }
    atomicAdd(&sum[t], s);
    atomicAdd(&sumsq[t], ss);
}

// BatchNorm (batch stats, biased var) + optional gelu; optional second copy
__global__ void bn_apply(const float* __restrict__ X, float* __restrict__ Y,
                         float* __restrict__ Y2,
                         const float* __restrict__ sum, const float* __restrict__ sumsq,
                         const float* __restrict__ g, const float* __restrict__ b,
                         int n, int H, int act) {
    const size_t total = (size_t)n * H;
    const float inv_n = 1.f / (float)n;
    for (size_t i = (size_t)blockIdx.x * blockDim.x + threadIdx.x; i < total;
         i += (size_t)gridDim.x * blockDim.x) {
        const int t = (int)(i % (size_t)H);
        const float mean = sum[t] * inv_n;
        const float var = sumsq[t] * inv_n - mean * mean;
        float y = g[t] * (X[i] - mean) * rsqrtf(var + EPSN) + b[t];
        if (act) y = gelu_f(y);
        Y[i] = y;
        if (Y2) Y2[i] = y;
    }
}

// LayerNorm over H per row (+ optional gelu, optional residual). blockDim == H.
__global__ void ln_rows(const float* __restrict__ X, float* __restrict__ Y,
                        const float* __restrict__ g, const float* __restrict__ b,
                        const float* __restrict__ res, int H, int act) {
    const size_t base = (size_t)blockIdx.x * H;
    const int t = threadIdx.x;
    const float v = X[base + t];
    __shared__ float s1[32], s2[32];
    float s = wave_sum32(v);
    float ss = wave_sum32(v * v);
    const int nw = blockDim.x >> 5;
    if ((t & 31) == 0) { s1[t >> 5] = s; s2[t >> 5] = ss; }
    __syncthreads();
    if (t == 0) {
        float a = 0.f, c = 0.f;
        for (int i = 0; i < nw; ++i) { a += s1[i]; c += s2[i]; }
        s1[0] = a; s2[0] = c;
    }
    __syncthreads();
    const float mean = s1[0] / (float)H;
    const float var = s2[0] / (float)H - mean * mean;
    float y = g[t] * (v - mean) * rsqrtf(var + EPSN) + b[t];
    if (act) y = gelu_f(y);
    if (res) y += res[base + t];
    Y[base + t] = y;
}

// segment_sum of node features into per-graph pool (batch sorted -> run flush)
__global__ void seg_sum_nodes(const float* __restrict__ X, const int* __restrict__ batch,
                              float* __restrict__ out, int n, int H) {
    const int t = threadIdx.x;        // blockDim.x == H
    const int r0 = blockIdx.x * 64;
    const int r1 = min(r0 + 64, n);
    float acc = 0.f; int cur = -1;
    for (int r = r0; r < r1; ++r) {
        const int gidx = batch[r];
        if (gidx != cur) {
            if (cur >= 0) atomicAdd(&out[(size_t)cur * H + t], acc);
            cur = gidx; acc = 0.f;
        }
        acc += X[(size_t)r * H + t];
    }
    if (cur >= 0) atomicAdd(&out[(size_t)cur * H + t], acc);
}

// x += vn[batch[row]]
__global__ void add_vn(float* __restrict__ x, const float* __restrict__ vn,
                       const int* __restrict__ batch, int n, int H) {
    const int row = blockIdx.x;
    if (row >= n) return;
    const int g = batch[row];
    x[(size_t)row * H + threadIdx.x] += vn[(size_t)g * H + threadIdx.x];
}

// h = (1 + eps) * x   (scatter destination init)
__global__ void h_init(const float* __restrict__ x, float* __restrict__ h,
                       const float* __restrict__ epsl, size_t total) {
    const float e = 1.f + epsl[0];
    for (size_t i = (size_t)blockIdx.x * blockDim.x + threadIdx.x; i < total;
         i += (size_t)gridDim.x * blockDim.x) h[i] = e * x[i];
}

// h[dst] += x[src] for every edge (float4 chunks, 64 chunks per edge when H=256)
__global__ void edge_scatter(const float* __restrict__ X, float* __restrict__ Hb,
                             const int* __restrict__ src, const int* __restrict__ dst,
                             int e, int H) {
    const int chunks = H >> 2;
    const size_t total = (size_t)e * chunks;
    for (size_t idx = (size_t)blockIdx.x * blockDim.x + threadIdx.x; idx < total;
         idx += (size_t)gridDim.x * blockDim.x) {
        const int edge = (int)(idx / chunks);
        const int c = (int)(idx % chunks);
        const int s = src[edge], d = dst[edge];
        const float4 v = ((const float4*)(X + (size_t)s * H))[c];
        float* p = Hb + (size_t)d * H + (size_t)c * 4;
        atomicAdd(p + 0, v.x);
        atomicAdd(p + 1, v.y);
        atomicAdd(p + 2, v.z);
        atomicAdd(p + 3, v.w);
    }
}

// x = sigmoid(gate) * o + (1 - sigmoid(gate)) * h
__global__ void combine_gate(const float* __restrict__ gate, const float* __restrict__ o,
                             const float* __restrict__ h, float* __restrict__ out, size_t total) {
    for (size_t i = (size_t)blockIdx.x * blockDim.x + threadIdx.x; i < total;
         i += (size_t)gridDim.x * blockDim.x) {
        const float gt = 1.f / (1.f + expf(-gate[i]));
        out[i] = gt * o[i] + (1.f - gt) * h[i];
    }
}

// logits[l, row] = dot(proj[l, row, :], jk_attn[l, :]).  blockIdx = l*n + row
__global__ void jk_logits(const float* __restrict__ proj, const float* __restrict__ attn,
                          float* __restrict__ logits, int n, int H) {
    const int id = blockIdx.x;
    const int l = id / n, row = id % n;
    const int t = threadIdx.x;
    const float v = proj[((size_t)l * n + row) * H + t] * attn[(size_t)l * H + t];
    __shared__ float s1[32];
    float s = wave_sum32(v);
    const int nw = blockDim.x >> 5;
    if ((t & 31) == 0) s1[t >> 5] = s;
    __syncthreads();
    if (t == 0) {
        float a = 0.f;
        for (int i = 0; i < nw; ++i) a += s1[i];
        logits[(size_t)l * n + row] = a;
    }
}

// softmax over layers + weighted sum: xj[row] = sum_l alpha[l,row] * proj[l,row]
__global__ void jk_mix(const float* __restrict__ proj, const float* __restrict__ logits,
                       float* __restrict__ xj, int n, int H, int L) {
    const int row = blockIdx.x;
    if (row >= n) return;
    const int t = threadIdx.x;
    float a[8];
    float mx = -3.4e38f;
    for (int l = 0; l < L; ++l) { a[l] = logits[(size_t)l * n + row]; mx = fmaxf(mx, a[l]); }
    float den = 0.f;
    for (int l = 0; l < L; ++l) { a[l] = expf(a[l] - mx); den += a[l]; }
    float o = 0.f;
    for (int l = 0; l < L; ++l) o += a[l] * proj[((size_t)l * n + row) * H + t];
    xj[(size_t)row * H + t] = o / den;
}

// per-graph multi-pool readout (batch is sorted; binary search the segment)
__global__ void pool_graphs(const float* __restrict__ xj, const int* __restrict__ batch,
                            const float* __restrict__ pool_w, float* __restrict__ pooled,
                            int n, int H) {
    const int g = blockIdx.x;
    const int t = threadIdx.x;
    __shared__ int lohi[2];
    if (t == 0) {
        int lo = 0, hb = n;
        while (lo < hb) { int m = (lo + hb) >> 1; if (batch[m] < g) lo = m + 1; else hb = m; }
        lohi[0] = lo;
        lo = 0; hb = n;
        while (lo < hb) { int m = (lo + hb) >> 1; if (batch[m] < g + 1) lo = m + 1; else hb = m; }
        lohi[1] = lo;
    }
    __syncthreads();
    const int lo = lohi[0], hi = lohi[1];
    float add = 0.f, mx = -3.4e38f;
    for (int r = lo; r < hi; ++r) {
        const float v = xj[(size_t)r * H + t];
        add += v; mx = fmaxf(mx, v);
    }
    const float cnt = (float)(hi - lo);
    const float w0 = pool_w[0], w1 = pool_w[1], w2 = pool_w[2];
    const float m = fmaxf(w0, fmaxf(w1, w2));
    const float e0 = expf(w0 - m), e1 = expf(w1 - m), e2 = expf(w2 - m);
    const float d = e0 + e1 + e2;
    pooled[(size_t)g * H + t] = (e0 / d) * add + (e1 / d) * (add / cnt) + (e2 / d) * mx;
}

// ---------------------------------------------------------------------------
extern "C" void kernel_launch(void* const* d_in, const int* in_sizes, int n_in,
                              void* d_out, int out_size, void* d_ws, size_t ws_size,
                              hipStream_t stream) {
    const float* x_in    = (const float*)d_in[0];
    const int*   edge    = (const int*)d_in[1];
    const int*   batch   = (const int*)d_in[2];
    const float* in_bn_g = (const float*)d_in[4];
    const float* in_bn_b = (const float*)d_in[5];
    const float* fc1_w   = (const float*)d_in[6];
    const float* fc1_b   = (const float*)d_in[7];
    const float* n1_g    = (const float*)d_in[8];
    const float* n1_b    = (const float*)d_in[9];
    const float* fc2_w   = (const float*)d_in[10];
    const float* fc2_b   = (const float*)d_in[11];
    const float* n2_g    = (const float*)d_in[12];
    const float* n2_b    = (const float*)d_in[13];
    const float* gate_w  = (const float*)d_in[14];
    const float* gate_b  = (const float*)d_in[15];
    const float* eps     = (const float*)d_in[16];
    const float* bn_g    = (const float*)d_in[17];
    const float* bn_b    = (const float*)d_in[18];
    const float* vn_w1   = (const float*)d_in[19];
    const float* vn_b1   = (const float*)d_in[20];
    const float* vn_w2   = (const float*)d_in[21];
    const float* vn_b2   = (const float*)d_in[22];
    const float* jk_w    = (const float*)d_in[23];
    const float* jk_b    = (const float*)d_in[24];
    const float* jk_attn = (const float*)d_in[25];
    const float* pool_w  = (const float*)d_in[26];
    const float* fco1_w  = (const float*)d_in[27];
    const float* fco1_b  = (const float*)d_in[28];
    const float* ln_g    = (const float*)d_in[29];
    const float* ln_b    = (const float*)d_in[30];
    const float* fco2_w  = (const float*)d_in[31];
    const float* fco2_b  = (const float*)d_in[32];

    const int H   = in_sizes[4];            // 256
    const int N   = in_sizes[0] / H;        // 20000
    const int E   = in_sizes[1] / 2;        // 320000
    const int L   = in_sizes[16];           // 4
    const int LAT = in_sizes[32];           // 128
    const int G   = out_size / LAT;         // 128
    const int* src = edge;
    const int* dst = edge + E;
    const size_t HH = (size_t)H * H;
    const size_t NF = (size_t)N * H;

    // workspace layout (floats)
    float* ws    = (float*)d_ws;
    float* xcur  = ws;
    float* hbuf  = ws + 1 * NF;
    float* gateb = ws + 2 * NF;
    float* t1    = ws + 3 * NF;
    float* t2    = ws + 4 * NF;
    float* hs    = ws + 5 * NF;             // L * NF
    float* proj  = hs + (size_t)L * NF;     // L * NF
    float* small = proj + (size_t)L * NF;
    float* bnsum = small;                   // H
    float* bnsq  = bnsum + H;               // H
    float* gpool = bnsq + H;                // G*H
    float* vn1   = gpool + (size_t)G * H;   // G*H
    float* vn2   = vn1 + (size_t)G * H;     // G*H
    float* lgts  = vn2 + (size_t)G * H;     // L*N
    float* pool  = lgts + (size_t)L * N;    // G*H
    float* oo    = pool + (size_t)G * H;    // G*H
    (void)ws_size; (void)n_in;

    auto gemm = [&](const float* A, const float* Wm, const float* bias, float* Cc,
                    int n, int K, int M, int act) {
        dim3 grid((unsigned)((n + 127) / 128), (unsigned)(M / 128));
        gemm_f32_wmma<<<grid, 256, 0, stream>>>(A, Wm, bias, Cc, n, K, M, act);
    };

    // ---- input BatchNorm -> xcur ----
    zero_f<<<64, 256, 0, stream>>>(bnsum, (size_t)2 * H);
    col_stats<<<128, H, 0, stream>>>(x_in, N, H, bnsum, bnsq);
    bn_apply<<<2048, 256, 0, stream>>>(x_in, xcur, nullptr, bnsum, bnsq,
                                       in_bn_g, in_bn_b, N, H, 0);

    const size_t ec = (size_t)E * (H >> 2);
    const unsigned eblocks = (unsigned)((ec + 255) / 256);

    for (int l = 0; l < L; ++l) {
        // virtual node: pool -> mlp -> broadcast
        zero_f<<<64, 256, 0, stream>>>(gpool, (size_t)G * H);
        seg_sum_nodes<<<(N + 63) / 64, H, 0, stream>>>(xcur, batch, gpool, N, H);
        gemm(gpool, vn_w1 + l * HH, vn_b1 + (size_t)l * H, vn1, G, H, H, 1);
        gemm(vn1, vn_w2 + l * HH, vn_b2 + (size_t)l * H, vn2, G, H, H, 0);
        add_vn<<<N, H, 0, stream>>>(xcur, vn2, batch, N, H);

        // GINConv: h = (1+eps)*x + scatter-add(x[src] -> dst)
        h_init<<<2048, 256, 0, stream>>>(xcur, hbuf, eps + l, NF);
        edge_scatter<<<eblocks, 256, 0, stream>>>(xcur, hbuf, src, dst, E, H);

        // gated residual MLP
        gemm(hbuf, gate_w + l * HH, gate_b + (size_t)l * H, gateb, N, H, H, 0);
        gemm(hbuf, fc1_w + l * HH, fc1_b + (size_t)l * H, t1, N, H, H, 0);
        ln_rows<<<N, H, 0, stream>>>(t1, t1, n1_g + (size_t)l * H, n1_b + (size_t)l * H,
                                     nullptr, H, 1);
        gemm(t1, fc2_w + l * HH, fc2_b + (size_t)l * H, t2, N, H, H, 0);
        ln_rows<<<N, H, 0, stream>>>(t2, t2, n2_g + (size_t)l * H, n2_b + (size_t)l * H,
                                     nullptr, H, 1);
        combine_gate<<<2048, 256, 0, stream>>>(gateb, t2, hbuf, t1, NF);

        // BatchNorm + gelu -> xcur and hs[l]
        zero_f<<<64, 256, 0, stream>>>(bnsum, (size_t)2 * H);
        col_stats<<<128, H, 0, stream>>>(t1, N, H, bnsum, bnsq);
        bn_apply<<<2048, 256, 0, stream>>>(t1, xcur, hs + (size_t)l * NF, bnsum, bnsq,
                                           bn_g + (size_t)l * H, bn_b + (size_t)l * H,
                                           N, H, 1);
    }

    // ---- jumping-knowledge attention ----
    for (int l = 0; l < L; ++l)
        gemm(hs + (size_t)l * NF, jk_w + l * HH, jk_b + (size_t)l * H,
             proj + (size_t)l * NF, N, H, H, 0);
    jk_logits<<<L * N, H, 0, stream>>>(proj, jk_attn, lgts, N, H);
    jk_mix<<<N, H, 0, stream>>>(proj, lgts, t1, N, H, L);   // t1 = xj

    // ---- multi-pool readout + head ----
    pool_graphs<<<G, H, 0, stream>>>(t1, batch, pool_w, pool, N, H);
    gemm(pool, fco1_w, fco1_b, oo, G, H, H, 0);
    ln_rows<<<G, H, 0, stream>>>(oo, vn1, ln_g, ln_b, pool, H, 1);  // gelu(ln(..)) + pooled
    gemm(vn1, fco2_w, fco2_b, (float*)d_out, G, H, LAT, 0);
}